// ImgFeatAggregator_77910706749862
// MI455X (gfx1250) — compile-verified
//
#include <hip/hip_runtime.h>
#include <hip/hip_bf16.h>

typedef __attribute__((ext_vector_type(2))) float v2f;
typedef __attribute__((ext_vector_type(8))) float v8f;

#define EPSV 1e-5f
#define NQ   1024
#define NCAM 6

#if defined(__gfx1250__) && __has_builtin(__builtin_amdgcn_wmma_f32_16x16x4_f32)
#define HAVE_WMMA_F32X4 1
#else
#define HAVE_WMMA_F32X4 0
#endif

// -------- projection: grid coords + mask (+ ref3d copy) ---------------------

__device__ __forceinline__ void emit_cam(int cam, int q, float X, float Y, float Z,
                                         float img_w, float img_h,
                                         float* __restrict__ ws_xy,
                                         float* __restrict__ out_mask) {
    bool  zm    = Z > EPSV;
    float denom = fmaxf(Z, EPSV);
    float gx = (X / denom / img_w - 0.5f) * 2.0f;
    float gy = (Y / denom / img_h - 0.5f) * 2.0f;
    bool  m  = zm && (gx > -1.0f) && (gx < 1.0f) && (gy > -1.0f) && (gy < 1.0f);
    int idx = cam * NQ + q;
    ws_xy[idx * 2 + 0] = gx;
    ws_xy[idx * 2 + 1] = gy;
    out_mask[q * NCAM + cam] = m ? 1.0f : 0.0f;
}

#if !HAVE_WMMA_F32X4
__device__ __forceinline__ float projc(const float* __restrict__ l2i, int cam, int i,
                                       float px, float py, float pz) {
    const float* m = l2i + cam * 16 + i * 4;
    return m[0] * px + m[1] * py + m[2] * pz + m[3];
}
#endif

__global__ __launch_bounds__(32) void proj_kernel(
    const float* __restrict__ refpts,   // [1024,3]
    const float* __restrict__ l2i,      // [6,4,4]
    const float* __restrict__ pcr,      // [6]
    const int*   __restrict__ img_h_p,
    const int*   __restrict__ img_w_p,
    float* __restrict__ ws_xy,          // [6*1024*2]
    float* __restrict__ out_ref3d,      // [3072]
    float* __restrict__ out_mask)       // [1024*6]
{
    int lane = threadIdx.x;             // wave32, one wave per block
    int qcol = lane & 15;
    int q    = blockIdx.x * 16 + qcol;

    // ref3d passthrough copy (grid covers 64*32 = 2048 threads, 3072 elems)
    int tid = blockIdx.x * 32 + lane;
    for (int i = tid; i < NQ * 3; i += 64 * 32) out_ref3d[i] = refpts[i];

    float img_h = (float)(*img_h_p);
    float img_w = (float)(*img_w_p);

    // world-space point for this lane's query column
    float lo0 = pcr[0], lo1 = pcr[1], lo2 = pcr[2];
    float px = refpts[q * 3 + 0] * (pcr[3] - lo0) + lo0;
    float py = refpts[q * 3 + 1] * (pcr[4] - lo1) + lo1;
    float pz = refpts[q * 3 + 2] * (pcr[5] - lo2) + lo2;

    bool hi = lane >= 16;
    int cbase = hi ? 2 : 0;

#if HAVE_WMMA_F32X4
    // B (4x16 KxN, f32): lane = column N (query); low half holds K=0,1; high half K=2,3
    v2f B;
    B[0] = hi ? pz : px;
    B[1] = hi ? 1.0f : py;

    // A (16x4 MxK): row M = cam*4 + i ; lanes 0-15 hold K=0,1 ; 16-31 hold K=2,3
    int M  = lane & 15;
    int i  = M & 3;
    int j0 = hi ? 2 : 0;
    int camA = M >> 2;                       // WMMA#1: cams 0..3
    v2f A1, A2;
    A1[0] = l2i[camA * 16 + i * 4 + j0];
    A1[1] = l2i[camA * 16 + i * 4 + j0 + 1];
    int camB = 4 + (M >> 2);                 // WMMA#2: cams 4,5 (6,7 -> zero rows)
    bool ok  = camB < NCAM;
    A2[0] = ok ? l2i[camB * 16 + i * 4 + j0]     : 0.0f;
    A2[1] = ok ? l2i[camB * 16 + i * 4 + j0 + 1] : 0.0f;

    v8f Z = {0.f,0.f,0.f,0.f,0.f,0.f,0.f,0.f};
    // D[M][N]: lane holds column N = lane%16; VGPR r -> row r (low half) / 8+r (high half)
    v8f d1 = __builtin_amdgcn_wmma_f32_16x16x4_f32(false, A1, false, B, (short)0, Z, false, false);
    v8f d2 = __builtin_amdgcn_wmma_f32_16x16x4_f32(false, A2, false, B, (short)0, Z, false, false);

    // low lanes see rows 0-7 (cams 0,1 / 4,5); high lanes rows 8-15 (cams 2,3 / zeros)
    emit_cam(cbase + 0, q, d1[0], d1[1], d1[2], img_w, img_h, ws_xy, out_mask);
    emit_cam(cbase + 1, q, d1[4], d1[5], d1[6], img_w, img_h, ws_xy, out_mask);
    if (!hi) {
        emit_cam(4, q, d2[0], d2[1], d2[2], img_w, img_h, ws_xy, out_mask);
        emit_cam(5, q, d2[4], d2[5], d2[6], img_w, img_h, ws_xy, out_mask);
    }
#else
    emit_cam(cbase + 0, q, projc(l2i, cbase + 0, 0, px, py, pz),
             projc(l2i, cbase + 0, 1, px, py, pz), projc(l2i, cbase + 0, 2, px, py, pz),
             img_w, img_h, ws_xy, out_mask);
    emit_cam(cbase + 1, q, projc(l2i, cbase + 1, 0, px, py, pz),
             projc(l2i, cbase + 1, 1, px, py, pz), projc(l2i, cbase + 1, 2, px, py, pz),
             img_w, img_h, ws_xy, out_mask);
    if (!hi) {
        emit_cam(4, q, projc(l2i, 4, 0, px, py, pz), projc(l2i, 4, 1, px, py, pz),
                 projc(l2i, 4, 2, px, py, pz), img_w, img_h, ws_xy, out_mask);
        emit_cam(5, q, projc(l2i, 5, 0, px, py, pz), projc(l2i, 5, 1, px, py, pz),
                 projc(l2i, 5, 2, px, py, pz), img_w, img_h, ws_xy, out_mask);
    }
#endif
}

// -------- bilinear gather over all (q, cam, level), 256 channels ------------

__global__ __launch_bounds__(96) void sample_kernel(
    const float* __restrict__ f0, const float* __restrict__ f1,
    const float* __restrict__ f2, const float* __restrict__ f3,
    const float* __restrict__ ws_xy,
    float* __restrict__ out_sf)         // [256,1024,6,1,4] flat
{
    int t    = threadIdx.x;             // 96 = 4 queries * 24 (cam,lvl) pairs
    int pair = t % 24;
    int qs   = t / 24;
    int q    = blockIdx.x * 4 + qs;
    int lvl  = pair & 3;
    int n    = pair >> 2;

    int H = (lvl == 0) ? 116 : (lvl == 1) ? 58 : (lvl == 2) ? 29 : 15;
    int W = (lvl == 0) ? 200 : (lvl == 1) ? 100 : (lvl == 2) ? 50 : 25;
    const float* f = (lvl == 0) ? f0 : (lvl == 1) ? f1 : (lvl == 2) ? f2 : f3;
    int HW = H * W;

    float gx = ws_xy[(n * NQ + q) * 2 + 0];
    float gy = ws_xy[(n * NQ + q) * 2 + 1];

    float Wf = (float)W, Hf = (float)H;
    float x = (gx + 1.0f) * 0.5f * Wf - 0.5f;
    float y = (gy + 1.0f) * 0.5f * Hf - 0.5f;
    float x0f = floorf(x), y0f = floorf(y);
    float x1f = x0f + 1.0f, y1f = y0f + 1.0f;
    float wx1 = x - x0f, wx0 = 1.0f - wx1;
    float wy1 = y - y0f, wy0 = 1.0f - wy1;
    bool vx0 = (x0f >= 0.0f) && (x0f <= Wf - 1.0f);
    bool vx1 = (x1f >= 0.0f) && (x1f <= Wf - 1.0f);
    bool vy0 = (y0f >= 0.0f) && (y0f <= Hf - 1.0f);
    bool vy1 = (y1f >= 0.0f) && (y1f <= Hf - 1.0f);
    int ix0 = (int)fminf(fmaxf(x0f, 0.0f), Wf - 1.0f);
    int ix1 = (int)fminf(fmaxf(x1f, 0.0f), Wf - 1.0f);
    int iy0 = (int)fminf(fmaxf(y0f, 0.0f), Hf - 1.0f);
    int iy1 = (int)fminf(fmaxf(y1f, 0.0f), Hf - 1.0f);
    // zeros padding == zero the weight, keep loads clamped in-bounds
    float w00 = (vx0 && vy0) ? wx0 * wy0 : 0.0f;
    float w01 = (vx1 && vy0) ? wx1 * wy0 : 0.0f;
    float w10 = (vx0 && vy1) ? wx0 * wy1 : 0.0f;
    float w11 = (vx1 && vy1) ? wx1 * wy1 : 0.0f;

    int o00 = iy0 * W + ix0, o01 = iy0 * W + ix1;
    int o10 = iy1 * W + ix0, o11 = iy1 * W + ix1;
    const float* base = f + (size_t)n * 256 * HW;
    float* outp = out_sf + (size_t)q * 24 + pair;

    for (int c = 0; c < 256; ++c) {
        const float* bc = base + (size_t)c * HW;
        int cp = (c + 6 < 256) ? c + 6 : 255;   // prefetch a few channels ahead
        __builtin_prefetch(base + (size_t)cp * HW + o00, 0, 0);
        __builtin_prefetch(base + (size_t)cp * HW + o10, 0, 0);
        float v = w00 * bc[o00] + w01 * bc[o01] + w10 * bc[o10] + w11 * bc[o11];
        outp[(size_t)c * 24576] = v;            // c*1024*24 + q*24 + n*4 + lvl
    }
}

extern "C" void kernel_launch(void* const* d_in, const int* in_sizes, int n_in,
                              void* d_out, int out_size, void* d_ws, size_t ws_size,
                              hipStream_t stream) {
    const float* f0     = (const float*)d_in[0];
    const float* f1     = (const float*)d_in[1];
    const float* f2     = (const float*)d_in[2];
    const float* f3     = (const float*)d_in[3];
    const float* refpts = (const float*)d_in[4];
    const float* l2i    = (const float*)d_in[5];
    const float* pcr    = (const float*)d_in[6];
    const int*   ih     = (const int*)d_in[7];
    const int*   iw     = (const int*)d_in[8];

    float* out      = (float*)d_out;
    float* out_ref  = out;                       // 3072
    float* out_sf   = out + 3072;                // 6,291,456
    float* out_mask = out + 3072 + 256 * NQ * NCAM * 4; // 6144
    float* ws_xy    = (float*)d_ws;              // 6*1024 float2 = 48 KB

    proj_kernel<<<NQ / 16, 32, 0, stream>>>(refpts, l2i, pcr, ih, iw,
                                            ws_xy, out_ref, out_mask);
    sample_kernel<<<NQ / 4, 96, 0, stream>>>(f0, f1, f2, f3, ws_xy, out_sf);
}